// EdgeConv_88450556494199
// MI455X (gfx1250) — compile-verified
//
#include <hip/hip_runtime.h>

typedef __attribute__((ext_vector_type(2))) float v2f;
typedef __attribute__((ext_vector_type(8))) float v8f;

#define B_   8
#define CIN  64
#define NPTS 2048
#define COUT 128
#define MID  32
#define KNN  20
#define NEG_SLOPE 0.2f
#define BN_EPS 1e-5f

#define CHUNK 128
#define LDSTRIDE 132   // 128 + 4 pad to stagger LDS banks

// -------------------------------------------------------------------------
// xx[b,n] = sum_c x[b,c,n]^2
__global__ void __launch_bounds__(256) xx_kernel(const float* __restrict__ x,
                                                 float* __restrict__ xx) {
  int i = blockIdx.x * blockDim.x + threadIdx.x;   // b*N + n
  if (i >= B_ * NPTS) return;
  int b = i / NPTS, n = i % NPTS;
  const float* xb = x + (size_t)b * CIN * NPTS + n;
  float s = 0.f;
  #pragma unroll
  for (int c = 0; c < CIN; ++c) { float v = xb[c * NPTS]; s += v * v; }
  xx[i] = s;
}

// -------------------------------------------------------------------------
// Fused Gram (fp32 WMMA) + top-K.  One wave per 16 rows of dist[b].
__global__ void __launch_bounds__(32) topk_kernel(const float* __restrict__ x,
                                                  const float* __restrict__ xx,
                                                  int* __restrict__ idx_out) {
  __shared__ float lds[16 * LDSTRIDE];
  int bid  = blockIdx.x;             // b*(N/16) + row-tile
  int b    = bid >> 7;               // N/16 == 128
  int n0   = (bid & 127) << 4;
  int lane = threadIdx.x;
  int half = lane >> 4;              // K-pair selector for 16x16x4 fragments
  int lm   = lane & 15;

  const float* xb  = x  + (size_t)b * CIN * NPTS;
  const float* xxb = xx + b * NPTS;

  // A fragments: rows n0..n0+15, all K (reused for every column tile)
  v2f afrag[16];
  #pragma unroll
  for (int kk = 0; kk < 16; ++kk) {
    int c = kk * 4 + 2 * half;
    afrag[kk].x = xb[c * NPTS + n0 + lm];
    afrag[kk].y = xb[(c + 1) * NPTS + n0 + lm];
  }
  float xxn[8];
  #pragma unroll
  for (int r = 0; r < 8; ++r) xxn[r] = xxb[n0 + r + 8 * half];

  float vals[KNN];
  int   inds[KNN];
  #pragma unroll
  for (int k = 0; k < KNN; ++k) { vals[k] = -__builtin_inff(); inds[k] = 0; }

  for (int chunk = 0; chunk < NPTS / CHUNK; ++chunk) {
    int mbase = chunk * CHUNK;
    for (int t = 0; t < CHUNK / 16; ++t) {
      int m0 = mbase + t * 16;
      v8f acc = {0.f, 0.f, 0.f, 0.f, 0.f, 0.f, 0.f, 0.f};
      #pragma unroll
      for (int kk = 0; kk < 16; ++kk) {
        int c = kk * 4 + 2 * half;
        v2f bfrag;
        bfrag.x = xb[c * NPTS + m0 + lm];
        bfrag.y = xb[(c + 1) * NPTS + m0 + lm];
        acc = __builtin_amdgcn_wmma_f32_16x16x4_f32(
                  false, afrag[kk], false, bfrag, (short)0, acc, false, false);
      }
      float xxm = xxb[m0 + lm];
      #pragma unroll
      for (int r = 0; r < 8; ++r) {
        float d = 2.f * acc[r] - xxn[r] - xxm;   // dist = 2*inner - xx_n - xx_m
        lds[(r + 8 * half) * LDSTRIDE + t * 16 + lm] = d;
      }
    }
    __syncthreads();
    if (lane < 16) {                 // per-row register top-20, stable on ties
      const float* row = &lds[lane * LDSTRIDE];
      for (int j = 0; j < CHUNK; ++j) {
        float v = row[j];
        if (v > vals[KNN - 1]) {
          vals[KNN - 1] = v;
          inds[KNN - 1] = mbase + j;
          #pragma unroll
          for (int k = KNN - 1; k > 0; --k) {
            if (vals[k] > vals[k - 1]) {
              float tv = vals[k]; vals[k] = vals[k - 1]; vals[k - 1] = tv;
              int   ti = inds[k]; inds[k] = inds[k - 1]; inds[k - 1] = ti;
            }
          }
        }
      }
    }
    __syncthreads();
  }
  if (lane < 16) {
    int* out = idx_out + ((size_t)b * NPTS + n0 + lane) * KNN;
    #pragma unroll
    for (int k = 0; k < KNN; ++k) out[k] = inds[k];
  }
}

// -------------------------------------------------------------------------
// P  = scale o (W1 @ x)          -> Ps[b][m][o]  (transposed for gather)
// C  = scale o ((W2-W1) @ x) + shift -> Cs[b][n][o]
__global__ void __launch_bounds__(32) proj_kernel(const float* __restrict__ x,
    const float* __restrict__ conv_w,
    const float* __restrict__ gamma, const float* __restrict__ beta,
    const float* __restrict__ mean,  const float* __restrict__ var,
    float* __restrict__ Ps, float* __restrict__ Cs) {
  int bid = blockIdx.x;                    // b*(8*128) + oi*128 + mi
  int b   = bid / (8 * 128);
  int rem = bid % (8 * 128);
  int o0  = (rem / 128) * 16;
  int m0  = (rem % 128) * 16;
  int lane = threadIdx.x, half = lane >> 4, lm = lane & 15;

  const float* xb = x + (size_t)b * CIN * NPTS;
  v8f acc1 = {0.f,0.f,0.f,0.f,0.f,0.f,0.f,0.f};
  v8f acc2 = {0.f,0.f,0.f,0.f,0.f,0.f,0.f,0.f};
  #pragma unroll
  for (int kk = 0; kk < 16; ++kk) {
    int c = kk * 4 + 2 * half;
    const float* wrow = conv_w + (o0 + lm) * (2 * CIN) + c;
    v2f a1; a1.x = wrow[0];        a1.y = wrow[1];
    v2f a2; a2.x = wrow[CIN] - a1.x; a2.y = wrow[CIN + 1] - a1.y;  // W2 - W1
    v2f bf; bf.x = xb[c * NPTS + m0 + lm]; bf.y = xb[(c + 1) * NPTS + m0 + lm];
    acc1 = __builtin_amdgcn_wmma_f32_16x16x4_f32(false, a1, false, bf,
                                                 (short)0, acc1, false, false);
    acc2 = __builtin_amdgcn_wmma_f32_16x16x4_f32(false, a2, false, bf,
                                                 (short)0, acc2, false, false);
  }
  float* Psb = Ps + (size_t)b * NPTS * COUT;
  float* Csb = Cs + (size_t)b * NPTS * COUT;
  int m = m0 + lm;
  #pragma unroll
  for (int r = 0; r < 8; ++r) {
    int o = o0 + r + 8 * half;
    float sc = gamma[o] * rsqrtf(var[o] + BN_EPS);
    float sh = beta[o] - mean[o] * sc;
    Psb[m * COUT + o] = sc * acc1[r];
    Csb[m * COUT + o] = sc * acc2[r] + sh;
  }
}

// -------------------------------------------------------------------------
// out0[b,o,n] = max_k lrelu(Ps[b][idx_k][o] + Cs[b][n][o])
__global__ void __launch_bounds__(COUT) gather_max_kernel(
    const float* __restrict__ Ps, const float* __restrict__ Cs,
    const int* __restrict__ idx, float* __restrict__ out) {
  __shared__ int sidx[KNN];
  int bid = blockIdx.x;                  // b*N + n
  int b = bid >> 11;
  int n = bid & (NPTS - 1);
  int o = threadIdx.x;
  if (o < KNN) sidx[o] = idx[(size_t)bid * KNN + o];
  __syncthreads();
  const float* Psb = Ps + (size_t)b * NPTS * COUT;
  float c0 = Cs[(size_t)b * NPTS * COUT + n * COUT + o];
  float m = -__builtin_inff();
  #pragma unroll
  for (int k = 0; k < KNN; ++k) {
    float v = Psb[sidx[k] * COUT + o] + c0;
    v = v > 0.f ? v : NEG_SLOPE * v;
    m = fmaxf(m, v);
  }
  out[((size_t)b * COUT + o) * NPTS + n] = m;
}

// -------------------------------------------------------------------------
__global__ void __launch_bounds__(256) mean_kernel(const float* __restrict__ out,
                                                   float* __restrict__ w) {
  __shared__ float red[256];
  const float* p = out + (size_t)blockIdx.x * NPTS;   // row = b*COUT + o
  float s = 0.f;
  for (int i = threadIdx.x; i < NPTS; i += 256) s += p[i];
  red[threadIdx.x] = s;
  __syncthreads();
  for (int st = 128; st > 0; st >>= 1) {
    if ((int)threadIdx.x < st) red[threadIdx.x] += red[threadIdx.x + st];
    __syncthreads();
  }
  if (threadIdx.x == 0) w[blockIdx.x] = red[0] * (1.f / NPTS);
}

// -------------------------------------------------------------------------
__global__ void __launch_bounds__(COUT) se_kernel(const float* __restrict__ w,
    const float* __restrict__ w1, const float* __restrict__ w2,
    float* __restrict__ g) {
  __shared__ float sw[COUT];
  __shared__ float sh[MID];
  int b = blockIdx.x, t = threadIdx.x;
  sw[t] = w[b * COUT + t];
  __syncthreads();
  if (t < MID) {
    float s = 0.f;
    #pragma unroll
    for (int o = 0; o < COUT; ++o) s += sw[o] * w1[t * COUT + o];
    sh[t] = s > 0.f ? s : 0.f;
  }
  __syncthreads();
  float s = 0.f;
  #pragma unroll
  for (int m = 0; m < MID; ++m) s += sh[m] * w2[t * MID + m];
  g[b * COUT + t] = 1.f / (1.f + expf(-s));
}

// -------------------------------------------------------------------------
__global__ void __launch_bounds__(256) scale_kernel(float* __restrict__ out,
                                                    const float* __restrict__ g) {
  int i = blockIdx.x * blockDim.x + threadIdx.x;
  if (i < B_ * COUT * NPTS) out[i] *= g[i >> 11];   // i / NPTS -> b*COUT+o
}

// -------------------------------------------------------------------------
extern "C" void kernel_launch(void* const* d_in, const int* in_sizes, int n_in,
                              void* d_out, int out_size, void* d_ws, size_t ws_size,
                              hipStream_t stream) {
  const float* x      = (const float*)d_in[0];
  const float* conv_w = (const float*)d_in[1];
  const float* gamma  = (const float*)d_in[2];
  const float* beta   = (const float*)d_in[3];
  const float* mean   = (const float*)d_in[4];
  const float* var    = (const float*)d_in[5];
  const float* se_w1  = (const float*)d_in[6];
  const float* se_w2  = (const float*)d_in[7];
  float* out = (float*)d_out;

  // workspace carve-up (~18.3 MB)
  float* xx   = (float*)d_ws;                         // B*N
  int*   idx  = (int*)(xx + B_ * NPTS);               // B*N*K
  float* Ps   = (float*)(idx + B_ * NPTS * KNN);      // B*N*COUT
  float* Cs   = Ps + (size_t)B_ * NPTS * COUT;        // B*N*COUT
  float* wacc = Cs + (size_t)B_ * NPTS * COUT;        // B*COUT
  float* g    = wacc + B_ * COUT;                     // B*COUT

  xx_kernel<<<(B_ * NPTS + 255) / 256, 256, 0, stream>>>(x, xx);
  topk_kernel<<<B_ * (NPTS / 16), 32, 0, stream>>>(x, xx, idx);
  proj_kernel<<<B_ * (COUT / 16) * (NPTS / 16), 32, 0, stream>>>(
      x, conv_w, gamma, beta, mean, var, Ps, Cs);
  gather_max_kernel<<<B_ * NPTS, COUT, 0, stream>>>(Ps, Cs, idx, out);
  mean_kernel<<<B_ * COUT, 256, 0, stream>>>(out, wacc);
  se_kernel<<<B_, COUT, 0, stream>>>(wacc, se_w1, se_w2, g);
  scale_kernel<<<(B_ * COUT * NPTS + 255) / 256, 256, 0, stream>>>(out, g);
}